// LocallyConnected2d_738734375233
// MI455X (gfx1250) — compile-verified
//
#include <hip/hip_runtime.h>
#include <stdint.h>

// LocallyConnected2d: x[64,32,64,64] (f32), weight[1,64,32,62,62,9] (f32)
// out[64,64,62,62] = per-(h,w) GEMM: M=batch(64), N=cout(64), K=cin*9(288)

#define BATCH 64
#define CIN   32
#define H_IN  64
#define W_IN  64
#define COUT  64
#define OH    62
#define OW    62
#define K2    9
#define KTOT  288   // CIN * K2
#define KC    96    // K-chunk staged in LDS (3 chunks)
#define KP    100   // padded LDS row stride in floats (16B-aligned, conflict-skewed)

typedef __attribute__((ext_vector_type(2))) float v2f;
typedef __attribute__((ext_vector_type(8))) float v8f;

// CDNA5 async global->LDS copy (VGLOBAL GLOBAL_LOAD_ASYNC_TO_LDS_B32, ASYNCcnt).
// lds_off: wave-relative LDS byte address (low 32 bits of generic LDS pointer);
// g: per-lane global address (GV mode, 64-bit vaddr).
__device__ __forceinline__ void async_copy_b32(uint32_t lds_off, const float* g) {
    asm volatile("global_load_async_to_lds_b32 %0, %1, off"
                 :: "v"(lds_off), "v"(g)
                 : "memory");
}

__device__ __forceinline__ void wait_async_all() {
    asm volatile("s_wait_asynccnt 0x0" ::: "memory");
}

__global__ __launch_bounds__(256)
void lc2d_wmma_f32_kernel(const float* __restrict__ x,
                          const float* __restrict__ w,
                          float* __restrict__ out)
{
    // A: patches [m=batch][k], B: weight transposed [n=cout][k]
    __shared__ float As[BATCH * KP];
    __shared__ float Bs[COUT * KP];

    // LDS aperture: low 32 bits of the generic pointer == wave-relative LDS addr
    const uint32_t asBase = (uint32_t)(uintptr_t)&As[0];
    const uint32_t bsBase = (uint32_t)(uintptr_t)&Bs[0];

    const int loc = blockIdx.x;          // spatial location index
    const int oh  = loc / OW;
    const int ow  = loc - oh * OW;

    const int tid  = threadIdx.x;
    const int wave = tid >> 5;           // 0..7 (wave32)
    const int lane = tid & 31;
    const int half = lane >> 4;          // 0: lanes 0-15, 1: lanes 16-31
    const int row  = lane & 15;

    // 4x4 grid of 16x16 output tiles; each wave owns two N-adjacent tiles
    const int tm  = wave >> 1;           // 0..3
    const int tn0 = (wave & 1) * 2;      // 0 or 2

    v8f acc0 = {};
    v8f acc1 = {};

    for (int kc = 0; kc < KTOT / KC; ++kc) {
        const int k0 = kc * KC;
        __syncthreads();   // LDS write-after-read protection across chunks

        // ---- Stage A (im2col patches) via async global->LDS: 64x96 dwords ----
        for (int e = tid; e < BATCH * KC; e += 256) {
            const int m  = e / KC;
            const int kk = e - m * KC;
            const int k  = k0 + kk;
            const int ci = k / K2;
            const int r  = k - ci * K2;
            const int di = r / 3;
            const int dj = r - di * 3;
            async_copy_b32(asBase + (uint32_t)(m * KP + kk) * 4u,
                           &x[(((m * CIN) + ci) * H_IN + (oh + di)) * W_IN + (ow + dj)]);
        }
        // ---- Stage B (per-location weights, transposed to [n][k]) ----
        for (int e = tid; e < COUT * KC; e += 256) {
            const int n  = e / KC;
            const int kk = e - n * KC;
            const int k  = k0 + kk;
            const int ci = k / K2;
            const int r  = k - ci * K2;
            async_copy_b32(bsBase + (uint32_t)(n * KP + kk) * 4u,
                           &w[(((n * CIN) + ci) * (OH * OW) + loc) * K2 + r]);
        }
        wait_async_all();   // this wave's LDS writes are complete
        __syncthreads();    // publish to all waves

        const float* aRow  = &As[(tm * 16 + row) * KP];
        const float* bRow0 = &Bs[(tn0 * 16 + row) * KP];
        const float* bRow1 = &Bs[((tn0 + 1) * 16 + row) * KP];

        // ---- 24 k-steps of V_WMMA_F32_16X16X4_F32 per chunk ----
        // A 16x4 layout: v0 = K(2*half), v1 = K(2*half+1) per lane (ISA 7.12.2)
        #pragma unroll
        for (int k = 0; k < KC; k += 4) {
            v2f a, b0, b1;
            a.x  = aRow[k + 2 * half];
            a.y  = aRow[k + 2 * half + 1];
            b0.x = bRow0[k + 2 * half];
            b0.y = bRow0[k + 2 * half + 1];
            b1.x = bRow1[k + 2 * half];
            b1.y = bRow1[k + 2 * half + 1];
            acc0 = __builtin_amdgcn_wmma_f32_16x16x4_f32(
                false, a, false, b0, (short)0, acc0, false, false);
            acc1 = __builtin_amdgcn_wmma_f32_16x16x4_f32(
                false, a, false, b1, (short)0, acc1, false, false);
        }
    }

    // ---- Store: C/D layout — VGPR j holds M = 8*half + j, N = row (ISA 7.12.2) ----
    #pragma unroll
    for (int j = 0; j < 8; ++j) {
        const int m  = tm * 16 + half * 8 + j;   // batch index
        const int n0 = tn0 * 16 + row;           // cout index (tile tn0)
        out[(m * COUT + n0) * (OH * OW) + loc]       = acc0[j];
        out[(m * COUT + n0 + 16) * (OH * OW) + loc]  = acc1[j];
    }
}

extern "C" void kernel_launch(void* const* d_in, const int* in_sizes, int n_in,
                              void* d_out, int out_size, void* d_ws, size_t ws_size,
                              hipStream_t stream) {
    const float* x   = (const float*)d_in[0];   // [64,32,64,64]
    const float* wgt = (const float*)d_in[1];   // [1,64,32,62,62,9]
    float* out       = (float*)d_out;           // [64,64,62,62]

    dim3 grid(OH * OW);   // 3844 blocks, one per output location
    dim3 block(256);      // 8 wave32 waves
    lc2d_wmma_f32_kernel<<<grid, block, 0, stream>>>(x, wgt, out);
}